// ConsistencyStopController2_19542101196886
// MI455X (gfx1250) — compile-verified
//
#include <hip/hip_runtime.h>
#include <hip/hip_bf16.h>
#include <math.h>

// Dims from the reference
#define T_DIM  512
#define S_DIM  256
#define TAU    64
#define B_DIM  10
#define A_DIM  4

typedef __attribute__((ext_vector_type(2))) float v2f;
typedef __attribute__((ext_vector_type(8))) float v8f;

// ---------------------------------------------------------------------------
// Kernel 1: t_i = s_i @ W_tau + b_tau          (512 x 64, K=256)
// grid = T_DIM blocks, block = 64 threads (one thread per tau column)
// ---------------------------------------------------------------------------
__global__ void tau_kernel(const float* __restrict__ s,
                           const float* __restrict__ W_tau,
                           const float* __restrict__ b_tau,
                           float* __restrict__ t) {
    const int i = blockIdx.x;
    const int k = threadIdx.x;            // 0..63
    const float* srow = s + i * S_DIM;
    float acc = b_tau[k];
    for (int kk = 0; kk < S_DIM; ++kk)
        acc = fmaf(srow[kk], W_tau[kk * TAU + k], acc);
    t[i * TAU + k] = acc;
}

// ---------------------------------------------------------------------------
// Kernel 2: per-row auxiliaries
//   alpha_base[r] = t[r] @ W_trans[:64] + b_trans         (512 x 64)
//   tnorm[r]      = ||t[r]||^2
//   H[r][b]       = dot(t[r], W_trans[64+b])              (512 x 10)
//   anorm2[r][b]  = ||alpha_base[r] + W_trans[64+b]||^2   (512 x 10)
// grid = T_DIM blocks, block = 64 threads
// ---------------------------------------------------------------------------
__global__ void aux_kernel(const float* __restrict__ t,
                           const float* __restrict__ W_trans,
                           const float* __restrict__ b_trans,
                           float* __restrict__ alpha_base,
                           float* __restrict__ tnorm,
                           float* __restrict__ H,
                           float* __restrict__ anorm2) {
    const int r = blockIdx.x;
    const int k = threadIdx.x;            // 0..63
    __shared__ float sh_t[TAU];
    __shared__ float sh_ab[TAU];

    sh_t[k] = t[r * TAU + k];
    __syncthreads();

    float ab = b_trans[k];
    for (int kk = 0; kk < TAU; ++kk)
        ab = fmaf(sh_t[kk], W_trans[kk * TAU + k], ab);
    alpha_base[r * TAU + k] = ab;
    sh_ab[k] = ab;
    __syncthreads();

    if (k == 0) {
        float s2 = 0.f;
        for (int kk = 0; kk < TAU; ++kk) s2 = fmaf(sh_t[kk], sh_t[kk], s2);
        tnorm[r] = s2;
    }
    if (k < B_DIM) {
        const int b = k;
        const float* w1 = W_trans + (TAU + b) * TAU;
        float h = 0.f, a2 = 0.f;
        for (int kk = 0; kk < TAU; ++kk) {
            float w = w1[kk];
            h  = fmaf(sh_t[kk], w, h);
            float a = sh_ab[kk] + w;
            a2 = fmaf(a, a, a2);
        }
        H[r * B_DIM + b]      = h;
        anorm2[r * B_DIM + b] = a2;
    }
}

// ---------------------------------------------------------------------------
// Kernel 3: heads (action_logps, start_logps) -- tiny
// grid = T_DIM blocks, block = 64 threads
// ---------------------------------------------------------------------------
__global__ void heads_kernel(const float* __restrict__ s,
                             const float* __restrict__ t,
                             const float* __restrict__ W_micro,
                             const float* __restrict__ b_micro,
                             const float* __restrict__ W_pol,
                             const float* __restrict__ b_pol,
                             float* __restrict__ out_action,
                             float* __restrict__ out_start) {
    const int i   = blockIdx.x;
    const int tid = threadIdx.x;          // 0..63
    __shared__ float sm[B_DIM * A_DIM];   // 40 micro logits
    __shared__ float sp[B_DIM];           // 10 policy logits

    if (tid < B_DIM * A_DIM) {
        const float* srow = s + i * S_DIM;
        float acc = b_micro[tid];
        for (int kk = 0; kk < S_DIM; ++kk)
            acc = fmaf(srow[kk], W_micro[kk * (B_DIM * A_DIM) + tid], acc);
        sm[tid] = acc;
    }
    if (tid < B_DIM) {
        const float* trow = t + i * TAU;
        float acc = b_pol[tid];
        for (int kk = 0; kk < TAU; ++kk)
            acc = fmaf(trow[kk], W_pol[kk * B_DIM + tid], acc);
        sp[tid] = acc;
    }
    __syncthreads();

    if (tid < B_DIM * A_DIM) {
        const int g = (tid >> 2) << 2;    // start of this option's group of 4
        float mx = fmaxf(fmaxf(sm[g], sm[g + 1]), fmaxf(sm[g + 2], sm[g + 3]));
        float sum = expf(sm[g] - mx) + expf(sm[g + 1] - mx)
                  + expf(sm[g + 2] - mx) + expf(sm[g + 3] - mx);
        out_action[i * (B_DIM * A_DIM) + tid] = sm[tid] - mx - logf(sum);
    }
    if (tid < B_DIM) {
        float mx = sp[0];
        for (int b = 1; b < B_DIM; ++b) mx = fmaxf(mx, sp[b]);
        float sum = 0.f;
        for (int b = 0; b < B_DIM; ++b) sum += expf(sp[b] - mx);
        out_start[i * B_DIM + tid] = sp[tid] - mx - logf(sum);
    }
}

// ---------------------------------------------------------------------------
// Kernel 4: stop_logps via FP32 WMMA.
//   G[i,j] = dot(alpha_base[i], t[j])  (512x512, K=64)
//   d[i,j,b] = 2*G + (2*H[j,b] - tnorm[j]) - anorm2[i,b]   (<= 0)
//   stop_logps[i,j,b] = {d, log1p(-exp(d))}
// One wave32 per 16x16 (i,j) tile; 16 chained v_wmma_f32_16x16x4_f32.
// Epilogue stores 5 x b128 (float4) per (i,j): base stride 80B = 5*16B aligned.
// block = 256 (8 waves), grid = 1024/8 = 128 blocks.
// ---------------------------------------------------------------------------
__global__ void __launch_bounds__(256) stop_kernel(
        const float* __restrict__ alpha_base,
        const float* __restrict__ t,
        const float* __restrict__ tnorm,
        const float* __restrict__ anorm2,
        const float* __restrict__ H,
        float* __restrict__ out_stop) {
    const int lane = threadIdx.x & 31;
    const int wave = threadIdx.x >> 5;
    const int tile = blockIdx.x * 8 + wave;     // 0..1023
    const int ti = tile >> 5;                   // i-tile: 0..31
    const int tj = tile & 31;                   // j-tile: 0..31
    const int i_base = ti * 16;
    const int j_base = tj * 16;

    // A 16x4 f32 frag: lane&15 = M row, lane>=16 holds K+2..K+3
    // B 4x16  f32 frag: lane&15 = N col, lane>=16 holds K+2..K+3
    const int m    = lane & 15;
    const int koff = (lane >> 4) * 2;
    const float* arow = alpha_base + (i_base + m) * TAU + koff;  // A rows = i
    const float* brow = t          + (j_base + m) * TAU + koff;  // B cols = j

    v8f c = {};
#pragma unroll
    for (int k0 = 0; k0 < TAU; k0 += 4) {
        v2f av = *(const v2f*)(arow + k0);
        v2f bv = *(const v2f*)(brow + k0);
        // D = A*B + C, full fp32 (V_WMMA_F32_16X16X4_F32)
        c = __builtin_amdgcn_wmma_f32_16x16x4_f32(
                /*neg_a=*/false, av, /*neg_b=*/false, bv,
                /*c_mod=*/(short)0, c, /*reuse_a=*/false, /*reuse_b=*/false);
    }

    // Epilogue. D layout: VGPR v, lanes 0-15 -> M=v, lanes 16-31 -> M=8+v; N=lane&15.
    const int j  = j_base + m;
    const float tn = tnorm[j];
    float c2[B_DIM];                       // 2*H[j,b] - ||t_j||^2, hoisted
#pragma unroll
    for (int b = 0; b < B_DIM; ++b) c2[b] = fmaf(2.0f, H[j * B_DIM + b], -tn);

    const int ioff = (lane >> 4) * 8;
#pragma unroll
    for (int v = 0; v < 8; ++v) {
        const int i = i_base + ioff + v;
        const float g2 = 2.0f * c[v];
        const float* an = anorm2 + i * B_DIM;
        float4* dst4 = (float4*)(out_stop + (size_t)(i * T_DIM + j) * (B_DIM * 2));
#pragma unroll
        for (int p = 0; p < B_DIM / 2; ++p) {
            float d0 = fminf(g2 + c2[2 * p]     - an[2 * p],     0.0f);
            float d1 = fminf(g2 + c2[2 * p + 1] - an[2 * p + 1], 0.0f);
            float om0 = log1pf(-__expf(d0));   // log(1 - exp(d)), stable near 0
            float om1 = log1pf(-__expf(d1));
            dst4[p] = make_float4(d0, om0, d1, om1);  // [STOP, CONT] x 2
        }
    }
}

// ---------------------------------------------------------------------------
// Launch
// ---------------------------------------------------------------------------
extern "C" void kernel_launch(void* const* d_in, const int* in_sizes, int n_in,
                              void* d_out, int out_size, void* d_ws, size_t ws_size,
                              hipStream_t stream) {
    const float* s       = (const float*)d_in[0];   // (512,256)
    const float* W_tau   = (const float*)d_in[1];   // (256,64)
    const float* b_tau   = (const float*)d_in[2];   // (64,)
    const float* W_micro = (const float*)d_in[3];   // (256,40)
    const float* b_micro = (const float*)d_in[4];   // (40,)
    const float* W_pol   = (const float*)d_in[5];   // (64,10)
    const float* b_pol   = (const float*)d_in[6];   // (10,)
    const float* W_trans = (const float*)d_in[7];   // (74,64)
    const float* b_trans = (const float*)d_in[8];   // (64,)

    // Outputs, concatenated flat in return order
    float* out_action = (float*)d_out;                                    // 20480
    float* out_stop   = out_action + T_DIM * B_DIM * A_DIM;               // 5242880
    float* out_start  = out_stop + (size_t)T_DIM * T_DIM * B_DIM * 2;     // 5120

    // Workspace (fully overwritten every call)
    float* ws         = (float*)d_ws;
    float* t          = ws;                               // 512*64
    float* alpha_base = t + T_DIM * TAU;                  // 512*64
    float* tnorm      = alpha_base + T_DIM * TAU;         // 512
    float* anorm2     = tnorm + T_DIM;                    // 512*10
    float* H          = anorm2 + T_DIM * B_DIM;           // 512*10

    tau_kernel  <<<T_DIM, 64, 0, stream>>>(s, W_tau, b_tau, t);
    aux_kernel  <<<T_DIM, 64, 0, stream>>>(t, W_trans, b_trans,
                                           alpha_base, tnorm, H, anorm2);
    heads_kernel<<<T_DIM, 64, 0, stream>>>(s, t, W_micro, b_micro, W_pol, b_pol,
                                           out_action, out_start);
    // 32x32 = 1024 tiles of 16x16, 8 waves per block -> 128 blocks
    stop_kernel <<<128, 256, 0, stream>>>(alpha_base, t, tnorm, anorm2, H, out_stop);
}